// MyGraphAttentionLayer_15375982920432
// MI455X (gfx1250) — compile-verified
//
#include <hip/hip_runtime.h>
#include <hip/hip_bf16.h>

typedef __attribute__((ext_vector_type(2))) float v2f;
typedef __attribute__((ext_vector_type(8))) float v8f;

#define IN_F   256
#define OUT_F  32
#define HEADS  4
#define HF     (OUT_F * HEADS)   // 128
#define NEG_SLOPE 0.2f
#define EPS_GAT 1e-9f

__device__ __forceinline__ float lrelu(float v) {
    return v > 0.0f ? v : NEG_SLOPE * v;
}

// monotonic float <-> unsigned encoding for atomicMax on floats (incl. negatives)
__device__ __forceinline__ unsigned f2mono(float f) {
    unsigned u = __float_as_uint(f);
    return (u & 0x80000000u) ? ~u : (u | 0x80000000u);
}
__device__ __forceinline__ float mono2f(unsigned u) {
    return __uint_as_float((u & 0x80000000u) ? (u & 0x7FFFFFFFu) : ~u);
}

// ---------------------------------------------------------------------------
// 0) init: zero out + denom, reset max slots
// ---------------------------------------------------------------------------
__global__ void init_kernel(float* __restrict__ out, float* __restrict__ denom,
                            unsigned* __restrict__ slots, int out_n, int denom_n) {
    int stride = gridDim.x * blockDim.x;
    for (int i = blockIdx.x * blockDim.x + threadIdx.x; i < out_n; i += stride)
        out[i] = 0.0f;
    for (int i = blockIdx.x * blockDim.x + threadIdx.x; i < denom_n; i += stride)
        denom[i] = 0.0f;
    if (blockIdx.x == 0 && threadIdx.x < 8)
        slots[threadIdx.x] = 0u;   // decodes below any real float under mono ordering
}

// ---------------------------------------------------------------------------
// 1) GEMM: h[N,128] = x[N,256] @ W[256,128] with V_WMMA_F32_16X16X4_F32
//    one block = one 16-row panel, 8 waves = 8 column tiles of 16
// ---------------------------------------------------------------------------
#define XS_STRIDE 260   // 260 % 64 == 4 -> conflict-free per-column LDS reads
__global__ __launch_bounds__(256) void gemm_wmma_f32(const float* __restrict__ x,
                                                     const float* __restrict__ W,
                                                     float* __restrict__ h) {
    __shared__ float xs[16 * XS_STRIDE];
    const int panel = blockIdx.x;        // N/16 panels
    const int tid   = threadIdx.x;

    // cooperative coalesced load of 16x256 panel of x
    const float* xp = x + (size_t)panel * 16 * IN_F;
    for (int i = tid; i < 16 * IN_F; i += 256) {
        int r = i >> 8;          // /256
        int c = i & (IN_F - 1);
        xs[r * XS_STRIDE + c] = xp[i];
    }
    __syncthreads();

    const int wave = tid >> 5;           // 0..7 -> column tile
    const int lane = tid & 31;
    const int half = lane >> 4;          // 0: K={0,1}, 1: K={2,3}
    const int lrow = lane & 15;
    const int n0   = wave * 16;

    v8f acc = {};
    #pragma unroll 4
    for (int kk = 0; kk < IN_F; kk += 4) {
        const int ka = kk + 2 * half;
        // A fragment: A[M=lrow, K=ka..ka+1]
        v2f a;
        a.x = xs[lrow * XS_STRIDE + ka];
        a.y = xs[lrow * XS_STRIDE + ka + 1];
        // B fragment: B[K=ka..ka+1, N=n0+lrow]
        v2f b;
        b.x = W[(size_t)ka       * HF + n0 + lrow];
        b.y = W[(size_t)(ka + 1) * HF + n0 + lrow];
        acc = __builtin_amdgcn_wmma_f32_16x16x4_f32(
            /*neg_a=*/false, a, /*neg_b=*/false, b,
            /*c_mod=*/(short)0, acc, /*reuse_a=*/false, /*reuse_b=*/false);
    }

    // C/D layout: VGPR r -> row (r + 8*half), col n0 + lrow
    float* hp = h + (size_t)panel * 16 * HF;
    #pragma unroll
    for (int r = 0; r < 8; ++r)
        hp[(size_t)(r + 8 * half) * HF + n0 + lrow] = acc[r];
}

// ---------------------------------------------------------------------------
// 2) per-node attention scores: s_src[n,h] = h[n,h,:]·a_src[h], same for dst
// ---------------------------------------------------------------------------
__global__ void node_scores(const float* __restrict__ h,
                            const float* __restrict__ a_src,
                            const float* __restrict__ a_dst,
                            float* __restrict__ ssrc, float* __restrict__ sdst,
                            int total /* N*HEADS */) {
    int i = blockIdx.x * blockDim.x + threadIdx.x;
    if (i >= total) return;
    int n  = i >> 2;
    int hh = i & 3;
    const float* hp = h + (size_t)n * HF + hh * OUT_F;
    const float* as = a_src + hh * OUT_F;
    const float* ad = a_dst + hh * OUT_F;
    float s1 = 0.0f, s2 = 0.0f;
    #pragma unroll
    for (int f = 0; f < OUT_F; ++f) {
        float v = hp[f];
        s1 = fmaf(v, as[f], s1);
        s2 = fmaf(v, ad[f], s2);
    }
    ssrc[i] = s1;
    sdst[i] = s2;
}

// ---------------------------------------------------------------------------
// 3) per-head max over nodes of s_src and s_dst (softmax shift constant)
// ---------------------------------------------------------------------------
__global__ __launch_bounds__(256) void max_reduce(const float* __restrict__ ssrc,
                                                  const float* __restrict__ sdst,
                                                  unsigned* __restrict__ slots, int N) {
    const int tid    = blockIdx.x * blockDim.x + threadIdx.x;
    const int stride = gridDim.x * blockDim.x;
    float ms[HEADS], md[HEADS];
    #pragma unroll
    for (int hh = 0; hh < HEADS; ++hh) { ms[hh] = -3.4e38f; md[hh] = -3.4e38f; }
    for (int n = tid; n < N; n += stride) {
        #pragma unroll
        for (int hh = 0; hh < HEADS; ++hh) {
            ms[hh] = fmaxf(ms[hh], ssrc[n * HEADS + hh]);
            md[hh] = fmaxf(md[hh], sdst[n * HEADS + hh]);
        }
    }
    // wave32 butterfly reduce
    #pragma unroll
    for (int off = 16; off > 0; off >>= 1) {
        #pragma unroll
        for (int hh = 0; hh < HEADS; ++hh) {
            ms[hh] = fmaxf(ms[hh], __shfl_xor(ms[hh], off, 32));
            md[hh] = fmaxf(md[hh], __shfl_xor(md[hh], off, 32));
        }
    }
    if ((threadIdx.x & 31) == 0) {
        #pragma unroll
        for (int hh = 0; hh < HEADS; ++hh) {
            atomicMax(&slots[hh],         f2mono(ms[hh]));
            atomicMax(&slots[HEADS + hh], f2mono(md[hh]));
        }
    }
}

// ---------------------------------------------------------------------------
// 4) edge pass 1: e = exp(lrelu(s_src[row]+s_dst[col]) - C[h]); denom[row]+=e
// ---------------------------------------------------------------------------
__global__ void edge_denom(const long long* __restrict__ ei,
                           const float* __restrict__ ssrc,
                           const float* __restrict__ sdst,
                           const unsigned* __restrict__ slots,
                           float* __restrict__ denom, int E) {
    int e = blockIdx.x * blockDim.x + threadIdx.x;
    if (e >= E) return;
    const long long row = ei[e];
    const long long col = ei[(size_t)E + e];
    #pragma unroll
    for (int hh = 0; hh < HEADS; ++hh) {
        float C = lrelu(mono2f(slots[hh]) + mono2f(slots[HEADS + hh]));
        float v = ssrc[row * HEADS + hh] + sdst[col * HEADS + hh];
        float w = expf(lrelu(v) - C);
        atomicAdd(&denom[row * HEADS + hh], w);
    }
}

// ---------------------------------------------------------------------------
// 5) edge pass 2: out[row] += (e/denom[row]) * h[col]; one wave per edge,
//    each lane owns one feature across 4 heads (coalesced 512B h[col] read)
// ---------------------------------------------------------------------------
__global__ __launch_bounds__(256) void edge_aggregate(const long long* __restrict__ ei,
                                                      const float* __restrict__ h,
                                                      const float* __restrict__ ssrc,
                                                      const float* __restrict__ sdst,
                                                      const unsigned* __restrict__ slots,
                                                      const float* __restrict__ denom,
                                                      float* __restrict__ out, int E) {
    const int gid  = blockIdx.x * blockDim.x + threadIdx.x;
    const int e    = gid >> 5;
    const int lane = gid & 31;
    if (e >= E) return;
    const long long row = ei[e];
    const long long col = ei[(size_t)E + e];

    float wv = 0.0f, dv = 1.0f;
    if (lane < HEADS) {
        float C = lrelu(mono2f(slots[lane]) + mono2f(slots[HEADS + lane]));
        float v = ssrc[row * HEADS + lane] + sdst[col * HEADS + lane];
        wv = expf(lrelu(v) - C);
        dv = denom[row * HEADS + lane];
    }

    const float* hc = h   + (size_t)col * HF;
    float*       op = out + (size_t)row * HF;
    #pragma unroll
    for (int hh = 0; hh < HEADS; ++hh) {
        float alpha = __shfl(wv, hh, 32) / (__shfl(dv, hh, 32) + EPS_GAT);
        atomicAdd(&op[hh * OUT_F + lane], alpha * hc[hh * OUT_F + lane]);
    }
}

// ---------------------------------------------------------------------------
extern "C" void kernel_launch(void* const* d_in, const int* in_sizes, int n_in,
                              void* d_out, int out_size, void* d_ws, size_t ws_size,
                              hipStream_t stream) {
    const float*     x     = (const float*)d_in[0];
    const long long* ei    = (const long long*)d_in[1];
    const float*     W     = (const float*)d_in[2];
    const float*     a_src = (const float*)d_in[3];
    const float*     a_dst = (const float*)d_in[4];
    float*           out   = (float*)d_out;

    const int N = in_sizes[0] / IN_F;   // 100000
    const int E = in_sizes[1] / 2;      // 1600000

    // workspace layout
    float*    h     = (float*)d_ws;                    // N*128
    float*    ssrc  = h    + (size_t)N * HF;           // N*4
    float*    sdst  = ssrc + (size_t)N * HEADS;        // N*4
    float*    denom = sdst + (size_t)N * HEADS;        // N*4
    unsigned* slots = (unsigned*)(denom + (size_t)N * HEADS);  // 8

    init_kernel<<<2048, 256, 0, stream>>>(out, denom, slots, N * HF, N * HEADS);

    gemm_wmma_f32<<<N / 16, 256, 0, stream>>>(x, W, h);

    node_scores<<<(N * HEADS + 255) / 256, 256, 0, stream>>>(h, a_src, a_dst, ssrc, sdst, N * HEADS);

    max_reduce<<<512, 256, 0, stream>>>(ssrc, sdst, slots, N);

    edge_denom<<<(E + 255) / 256, 256, 0, stream>>>(ei, ssrc, sdst, slots, denom, E);

    {
        long long threads = (long long)E * 32;
        int blocks = (int)((threads + 255) / 256);
        edge_aggregate<<<blocks, 256, 0, stream>>>(ei, h, ssrc, sdst, slots, denom, out, E);
    }
}